// BoundaryMaxPooling_34943853920477
// MI455X (gfx1250) — compile-verified
//
#include <hip/hip_runtime.h>
#include <stdint.h>

// Problem constants (from the reference): B=4, C=512, T=256, K=504
#define BDIM 4
#define CDIM 512
#define TDIM 256
#define KDIM 504
#define LOG2T 8          // T = 256 = 2^8
#define NTHREADS 256

// ---------------------------------------------------------------------------
// CDNA5 async global->LDS copy helpers (ASYNCcnt-tracked).
// Builtin signatures (from hipcc diagnostics): global AS1 typed ptr first,
// then LDS AS3 typed ptr, then imm offset, imm cpol.
// ---------------------------------------------------------------------------
typedef int v4i_t __attribute__((ext_vector_type(4)));
typedef __attribute__((address_space(1))) int   as1_int;
typedef __attribute__((address_space(3))) int   as3_int;
typedef __attribute__((address_space(1))) v4i_t as1_v4i;
typedef __attribute__((address_space(3))) v4i_t as3_v4i;

#if __has_builtin(__builtin_amdgcn_global_load_async_to_lds_b32) && \
    __has_builtin(__builtin_amdgcn_global_load_async_to_lds_b128)
#define ASYNC_LOAD_B32(gsrc, ldst)                                   \
    __builtin_amdgcn_global_load_async_to_lds_b32(                   \
        (as1_int*)(uintptr_t)(gsrc),                                 \
        (as3_int*)(uint32_t)(uintptr_t)(ldst), 0, 0)
#define ASYNC_LOAD_B128(gsrc, ldst)                                  \
    __builtin_amdgcn_global_load_async_to_lds_b128(                  \
        (as1_v4i*)(uintptr_t)(gsrc),                                 \
        (as3_v4i*)(uint32_t)(uintptr_t)(ldst), 0, 0)
#else
#define ASYNC_LOAD_B32(gsrc, ldst)                                              \
    asm volatile("global_load_async_to_lds_b32 %0, %1, off"                     \
                 :: "v"((uint32_t)(uintptr_t)(ldst)), "v"((const void*)(gsrc))  \
                 : "memory")
#define ASYNC_LOAD_B128(gsrc, ldst)                                             \
    asm volatile("global_load_async_to_lds_b128 %0, %1, off"                    \
                 :: "v"((uint32_t)(uintptr_t)(ldst)), "v"((const void*)(gsrc))  \
                 : "memory")
#endif

#if __has_builtin(__builtin_amdgcn_s_wait_asynccnt)
#define WAIT_ASYNC0() __builtin_amdgcn_s_wait_asynccnt(0)
#else
#define WAIT_ASYNC0() asm volatile("s_wait_asynccnt 0x0" ::: "memory")
#endif

// ---------------------------------------------------------------------------
// One block per (b, c) row. Stage x[b,c,:] and segments[b] into LDS with async
// copies, build a 9-level sparse (RMQ) table in LDS, answer all K queries O(1).
// ---------------------------------------------------------------------------
__global__ __launch_bounds__(NTHREADS)
void boundary_max_pool_kernel(const float* __restrict__ x,
                              const float* __restrict__ seg,
                              float* __restrict__ out) {
    __shared__ float tab[(LOG2T + 1) * TDIM];  // sparse table: 9 x 256 floats
    __shared__ float segL[KDIM * 2];           // segment boundaries: 1008 floats

    const int c   = blockIdx.x;
    const int b   = blockIdx.y;
    const int tid = threadIdx.x;

    // --- Stage 1: async copies into LDS (tracked by ASYNCcnt) ---
    const float* xrow = x + ((size_t)(b * CDIM + c)) * TDIM;
    // 256 lanes x b32  -> 1024 B: the whole time row into table level 0
    ASYNC_LOAD_B32(xrow + tid, &tab[tid]);
    // 252 lanes x b128 -> 4032 B: this batch's (start,end) pairs
    if (tid < (KDIM * 2) / 4) {
        ASYNC_LOAD_B128(seg + (size_t)b * (KDIM * 2) + tid * 4, &segL[tid * 4]);
    }
    WAIT_ASYNC0();
    __syncthreads();

    // --- Stage 2: build sparse table, level l covers windows of 2^l ---
    // tab[l][t] = max(x[t .. min(t + 2^l - 1, T-1)])
#pragma unroll
    for (int l = 1; l <= LOG2T; ++l) {
        const int off = 1 << (l - 1);
        int t2 = tid + off;
        if (t2 > TDIM - 1) t2 = TDIM - 1;
        const float v = fmaxf(tab[(l - 1) * TDIM + tid],
                              tab[(l - 1) * TDIM + t2]);
        tab[l * TDIM + tid] = v;   // distinct row: no RAW hazard vs level l-1
        __syncthreads();
    }

    // --- Stage 3: O(1) range-max per segment ---
    float* outrow = out + ((size_t)(b * CDIM + c)) * KDIM;
#pragma unroll
    for (int k = tid; k < KDIM; k += NTHREADS) {
        const float st = segL[2 * k];
        const float en = segL[2 * k + 1];
        int s = (int)fminf(fmaxf(floorf(st), 0.0f), (float)(TDIM - 1));
        int e = (int)fminf(fmaxf(floorf(en), 0.0f), (float)(TDIM - 1));
        if (e < s) e = s;                      // at least one valid timestep
        const int len = e - s + 1;             // 1..256
        const int l   = 31 - __clz(len);       // floor(log2(len))
        const int off = 1 << l;
        outrow[k] = fmaxf(tab[l * TDIM + s], tab[l * TDIM + (e - off + 1)]);
    }
}

extern "C" void kernel_launch(void* const* d_in, const int* in_sizes, int n_in,
                              void* d_out, int out_size, void* d_ws, size_t ws_size,
                              hipStream_t stream) {
    (void)in_sizes; (void)n_in; (void)out_size; (void)d_ws; (void)ws_size;
    const float* x   = (const float*)d_in[0];   // [B, C, T] float32
    const float* seg = (const float*)d_in[1];   // [B, K, 2] float32
    float*       out = (float*)d_out;           // [B, C, K] float32

    dim3 grid(CDIM, BDIM);                      // one block per (b, c) row
    dim3 block(NTHREADS);
    boundary_max_pool_kernel<<<grid, block, 0, stream>>>(x, seg, out);
}